// ConstrainedDecoder_56349970923610
// MI455X (gfx1250) — compile-verified
//
#include <hip/hip_runtime.h>
#include <stdint.h>

// ---------------------------------------------------------------------------
// Constrained Viterbi decode, MI455X (gfx1250).
//   B=64 chains, T=2048 steps, N=32 tags. Sequential in t => latency-hiding
//   problem. TDM (tensor_load_to_lds) streams 16KB chunks into a 14-deep LDS
//   ring per chain, tracked with TENSORcnt. Max-plus semiring => no WMMA.
// ---------------------------------------------------------------------------

#define BATCH   64
#define TLEN    2048
#define NTAG    32
#define NINF    (-1.0e4f)

#define CHUNK        4                       // time steps per TDM descriptor
#define NCHUNK       14                      // ring depth (descriptors in flight)
#define STEP_FLOATS  (NTAG * NTAG)           // 1024 floats per step
#define STEP_BYTES   (STEP_FLOATS * 4)       // 4096 B
#define CHUNK_BYTES  (CHUNK * STEP_BYTES)    // 16384 B
#define RING_BYTES   (NCHUNK * CHUNK_BYTES)  // 229376 B
#define BP_BYTES     (TLEN * NTAG)           // 65536 B backpointers
#define SMEM_BYTES   (RING_BYTES + BP_BYTES + 128)   // 295040 <= 320KB/WGP

typedef uint32_t v4u __attribute__((ext_vector_type(4)));
typedef int32_t  v4i __attribute__((ext_vector_type(4)));
typedef int32_t  v8i __attribute__((ext_vector_type(8)));

#if defined(__has_builtin)
#  if __has_builtin(__builtin_amdgcn_tensor_load_to_lds)
#    define HAVE_TDM 1
#  endif
#  if __has_builtin(__builtin_amdgcn_s_wait_tensorcnt)
#    define WAIT_TENSOR(n) do { __builtin_amdgcn_s_wait_tensorcnt(n); \
                                asm volatile("" ::: "memory"); } while (0)
#  endif
#endif
#ifndef WAIT_TENSOR
#  define WAIT_TENSOR(n) asm volatile("s_wait_tensorcnt %0" :: "i"(n) : "memory")
#endif

// Issue one TDM descriptor: copy CHUNK steps (CHUNK x 1024 contiguous f32,
// viewed as a 2D tile of CHUNK rows x 1024 cols, row stride 1024) into LDS.
__device__ __forceinline__ void tdm_issue(const float* gsrc, float* ldst)
{
#if defined(HAVE_TDM)
    const uint64_t ga      = (uint64_t)(uintptr_t)gsrc;
    const uint32_t lds_off = (uint32_t)(uintptr_t)ldst;   // low 32 bits of pointer = LDS byte address

    // D# group 0: count=1 | lds_addr | global_addr[56:0] | type=2 ("image")
    v4u g0;
    g0[0] = 1u;
    g0[1] = lds_off;
    g0[2] = (uint32_t)ga;
    g0[3] = ((uint32_t)(ga >> 32) & 0x01FFFFFFu) | (2u << 30);

    // D# group 1: wg_mask=0 | data_size=4B | tensor_dim0=1024 | tensor_dim1=TLEN
    //             tile_dim0=1024 | tile_dim1=CHUNK | tile_dim2=0 | dim0_stride=1024
    v8i g1;
    g1[0] = (int)(2u << 16);                                    // data_size code 2
    g1[1] = (int)(((uint32_t)STEP_FLOATS & 0xFFFFu) << 16);     // dim0[15:0]
    g1[2] = (int)(((uint32_t)STEP_FLOATS >> 16)                 // dim0[31:16]
                | (((uint32_t)TLEN & 0xFFFFu) << 16));          // dim1[15:0]
    g1[3] = (int)(((uint32_t)TLEN >> 16)                        // dim1[31:16]
                | (((uint32_t)STEP_FLOATS & 0xFFFFu) << 16));   // tile_dim0
    g1[4] = (int)(CHUNK & 0xFFFF);                              // tile_dim1 (tile_dim2=0)
    g1[5] = (int)STEP_FLOATS;                                   // dim0_stride[31:0]
    g1[6] = 0;                                                  // stride hi | dim1_stride lo
    g1[7] = 0;
    v4i z4 = {0, 0, 0, 0};                   // D# groups 2/3 unused (2D tile)
    v8i z8 = {0, 0, 0, 0, 0, 0, 0, 0};       // extra group (clang-23 6-arg form)
    __builtin_amdgcn_tensor_load_to_lds(g0, g1, z4, z4, z8, 0);
    asm volatile("" ::: "memory");
#else
    // Portable synchronous fallback (keeps kernel functional without TDM builtin)
    const int lane = (int)(threadIdx.x & 31u);
    for (int k = lane; k < CHUNK * STEP_FLOATS; k += 32) ldst[k] = gsrc[k];
#endif
}

__global__ __launch_bounds__(32)
void viterbi_cdna5_kernel(const float* __restrict__ lp,
                          const uint8_t* __restrict__ maskp,
                          const uint8_t* __restrict__ scp,
                          const uint8_t* __restrict__ ecp,
                          const uint8_t* __restrict__ tcp,
                          float* __restrict__ outp)
{
    extern __shared__ unsigned char smem[];
    float*         ring   = (float*)smem;                      // 229376 B
    unsigned char* bp     = smem + RING_BYTES;                 //  65536 B
    float*         alphaL = (float*)(smem + RING_BYTES + BP_BYTES);

    const int lane = (int)threadIdx.x;           // lane j owns tag j
    const int b    = (int)blockIdx.x;            // one workgroup per batch row
    const float* lpb = lp + (size_t)b * TLEN * STEP_FLOATS;

    // ---- sequence length: mask is prefix-true, bytes are 0/1 -> popcount ----
    int len = 0;
    {
        const unsigned long long* m8 =
            (const unsigned long long*)(maskp + (size_t)b * TLEN);
        for (int k = 0; k < TLEN / 8; ++k) len += __popcll(m8[k]);
    }
    if (len < 1) len = 1;
    const int nchunks = (len + CHUNK - 1) / CHUNK;

    // ---- prologue: fill the TDM ring ----
    const int npro = (nchunks < NCHUNK) ? nchunks : NCHUNK;
    for (int c = 0; c < npro; ++c)
        tdm_issue(lpb + (size_t)c * CHUNK * STEP_FLOATS,
                  ring + (size_t)(c % NCHUNK) * (CHUNK * STEP_FLOATS));

    // ---- constraints (loop-invariant over t; penv lives in 32 VGPRs) ----
    const float startp = scp[lane] ? 0.0f : NINF;
    const float endp   = ecp[lane] ? 0.0f : NINF;
    float penv[NTAG];
    #pragma unroll
    for (int i = 0; i < NTAG; ++i)
        penv[i] = tcp[i * NTAG + lane] ? 0.0f : NINF;

    // ---- forward scan ----
    float alpha = 0.0f;
    for (int c = 0; c < nchunks; ++c) {
        // In-order TDM completion: <=NCHUNK-1 outstanding <=> chunk c landed.
        if (c + NCHUNK < nchunks) { WAIT_TENSOR(NCHUNK - 1); }
        else                      { WAIT_TENSOR(0); }          // pipeline tail

        const float* cb  = ring + (size_t)(c % NCHUNK) * (CHUNK * STEP_FLOATS);
        const int t0     = c * CHUNK;
        const int tend   = (t0 + CHUNK < len) ? (t0 + CHUNK) : len;

        for (int t = t0; t < tend; ++t) {
            const float* ph = cb + (size_t)(t - t0) * STEP_FLOATS;
            if (t == 0) {
                // alpha0[j] = max_i phi0[i][j] + start_pen[j] (+ end if len==1)
                float m = -3.0e38f;
                #pragma unroll
                for (int i = 0; i < NTAG; ++i)
                    m = fmaxf(m, ph[i * NTAG + lane]);
                alpha = m + startp;
                if (len == 1) alpha += endp;
            } else {
                float m = -3.0e38f;
                int   arg = 0;
                #pragma unroll
                for (int i = 0; i < NTAG; ++i) {
                    // broadcast alpha[i]: v_readlane -> SGPR, folds into v_add
                    float ai = __uint_as_float((unsigned)__builtin_amdgcn_readlane(
                                   (int)__float_as_uint(alpha), i));
                    float s  = (ph[i * NTAG + lane] + penv[i]) + ai;
                    bool  g  = s > m;          // strict '>' == first-max argmax
                    arg = g ? i : arg;
                    m   = g ? s : m;
                }
                if (t == len - 1) m += endp;   // column-additive: post-max OK
                alpha = m;
                bp[t * NTAG + lane] = (unsigned char)arg;
            }
        }
        // refill the slot we just consumed
        const int ci = c + NCHUNK;
        if (ci < nchunks)
            tdm_issue(lpb + (size_t)ci * CHUNK * STEP_FLOATS,
                      ring + (size_t)(ci % NCHUNK) * (CHUNK * STEP_FLOATS));
    }

    // ---- terminal max / argmax over the 32 lanes (uniform redundant scan) ----
    alphaL[lane] = alpha;
    __syncthreads();
    float best = -3.0e38f;
    int   btag = 0;
    for (int j = 0; j < NTAG; ++j) {
        float v = alphaL[j];
        if (v > best) { best = v; btag = j; }
    }
    if (lane == 0) outp[b] = best;

    // ---- backtrack ----
    int* tout = (int*)(outp + BATCH) + (size_t)b * TLEN;
    for (int t = len + lane; t < TLEN; t += NTAG) tout[t] = -1;   // padding

    int cur = btag;
    #pragma unroll 4
    for (int t = len - 1; t >= 1; --t) {
        // lane k preloads bp[t][k] (address independent of the path), then the
        // serial dependence is just a v_readlane, not an LDS round trip.
        unsigned rowv = (unsigned)bp[t * NTAG + lane];
        if (lane == 0) tout[t] = cur;
        cur = (int)((unsigned)__builtin_amdgcn_readlane((int)rowv, cur) & 0xFFu);
    }
    if (lane == 0) tout[0] = cur;
}

extern "C" void kernel_launch(void* const* d_in, const int* in_sizes, int n_in,
                              void* d_out, int out_size, void* d_ws, size_t ws_size,
                              hipStream_t stream)
{
    (void)in_sizes; (void)n_in; (void)out_size; (void)d_ws; (void)ws_size;
    const float*   lp    = (const float*)d_in[0];
    const uint8_t* maskp = (const uint8_t*)d_in[1];
    const uint8_t* scp   = (const uint8_t*)d_in[2];
    const uint8_t* ecp   = (const uint8_t*)d_in[3];
    const uint8_t* tcp   = (const uint8_t*)d_in[4];
    float*         outp  = (float*)d_out;   // [B] f32 probs, then [B*T] i32 tags

    (void)hipFuncSetAttribute((const void*)viterbi_cdna5_kernel,
                              hipFuncAttributeMaxDynamicSharedMemorySize,
                              SMEM_BYTES);
    viterbi_cdna5_kernel<<<dim3(BATCH), dim3(32), SMEM_BYTES, stream>>>(
        lp, maskp, scp, ecp, tcp, outp);
}